// OptimizedGNN_19155554140465
// MI455X (gfx1250) — compile-verified
//
#include <hip/hip_runtime.h>
#include <hip/hip_bf16.h>

// CDNA5 (gfx1250) GraphSAGE 3-layer forward.
// Bottleneck is edge scatter (~5 GB HBM traffic @ 23.3 TB/s); GEMMs run on
// V_WMMA_F32_16X16X4_F32 (exact fp32) since matrix FLOPs are negligible and
// all feature matrices are L2-resident (51 MB << 192 MB L2).

typedef __attribute__((ext_vector_type(2))) float v2f;
typedef __attribute__((ext_vector_type(8))) float v8f;

// Only meaningful in the device pass; host pass doesn't expose amdgcn builtins
// to __has_builtin (they are aux-target builtins there).
#if defined(__HIP_DEVICE_COMPILE__) && !__has_builtin(__builtin_amdgcn_wmma_f32_16x16x4_f32)
#error "wmma f32 16x16x4 builtin missing on this toolchain (device pass)"
#endif

// ---------------- utility fills ----------------
__global__ void fill_zero_f32(float* __restrict__ p, long n) {
    long i = (long)blockIdx.x * blockDim.x + threadIdx.x;
    if (i < n) p[i] = 0.0f;
}

__global__ void fill_zero_i32(int* __restrict__ p, int n) {
    int i = blockIdx.x * blockDim.x + threadIdx.x;
    if (i < n) p[i] = 0;
}

// ---------------- degree / inverse-degree ----------------
__global__ void degree_kernel(const int* __restrict__ dst, int* __restrict__ cnt, int E) {
    int e = blockIdx.x * blockDim.x + threadIdx.x;
    if (e < E) atomicAdd(&cnt[dst[e]], 1);
}

__global__ void invdeg_kernel(const int* __restrict__ cnt, float* __restrict__ inv, int N) {
    int i = blockIdx.x * blockDim.x + threadIdx.x;
    if (i < N) {
        int c = cnt[i];
        inv[i] = (c > 0) ? (1.0f / (float)c) : 0.0f;
    }
}

// ---------------- edge scatter: agg[dst] += h[src], C = 128 ----------------
// One wave (32 lanes) per edge; lane moves 4 consecutive floats (float4 load,
// 4x global_atomic_add_f32). Gather reads are 128B-coalesced per wave.
__global__ __launch_bounds__(256)
void scatter_add128(const float* __restrict__ h,
                    const int* __restrict__ src,
                    const int* __restrict__ dst,
                    float* __restrict__ agg, int E) {
    int t = blockIdx.x * blockDim.x + threadIdx.x;
    int e = t >> 5;
    if (e >= E) return;
    int lane = t & 31;
    int s = src[e];
    int d = dst[e];
    const float4 v = ((const float4*)(h + ((size_t)s << 7)))[lane];
    float* p = agg + ((size_t)d << 7) + (lane << 2);
    atomicAdd(p + 0, v.x);
    atomicAdd(p + 1, v.y);
    atomicAdd(p + 2, v.z);
    atomicAdd(p + 3, v.w);
}

// ---------------- fused SAGE GEMM epilogue kernel ----------------
// out[m][n] = act( sum_k agg[m][k]*inv[m]*Wl[n][k] + sum_k h[m][k]*Wr[n][k] + b[n] )
// Wave computes a 16x16 tile via V_WMMA_F32_16X16X4_F32, K in steps of 4.
// Requires N % 16 == 0 (100000 = 6250*16), Cout a multiple of 16.
// A fragment (16x4 f32): lane L -> row L&15, VGPR0/1 = K = (L>>4)*2 + {0,1}.
// B fragment (4x16 f32): lane L -> col L&15, VGPR0/1 = K = (L>>4)*2 + {0,1}.
// C/D fragment: VGPR r -> row r + 8*(L>>4), col L&15.
template <bool RELU>
__global__ __launch_bounds__(128)
void sage_gemm_wmma(const float* __restrict__ h,
                    const float* __restrict__ agg,
                    const float* __restrict__ inv,
                    const float* __restrict__ Wl,   // [Cout, K] row-major
                    const float* __restrict__ Wr,   // [Cout, K] row-major
                    const float* __restrict__ bias, // [Cout]
                    float* __restrict__ out,        // [N, Cout]
                    int K, int Cout) {
    const int lane = threadIdx.x & 31;
    const int wave = threadIdx.x >> 5;
    const int m0 = blockIdx.x << 4;
    const int n0 = ((blockIdx.y << 2) + wave) << 4;
    const int row = m0 + (lane & 15);
    const int col = n0 + (lane & 15);
    const int kh  = (lane >> 4) << 1;  // 0 or 2: K sub-offset for this half-wave

    const float* hrow = h   + (size_t)row * K + kh;
    const float* arow = agg + (size_t)row * K + kh;
    const float* wr   = Wr  + (size_t)col * K + kh;
    const float* wl   = Wl  + (size_t)col * K + kh;
    const float s = inv[row];

    v8f acc = {};

    // root term: h @ Wr^T
#pragma unroll 8
    for (int k = 0; k < K; k += 4) {
        v2f a = *(const v2f*)(hrow + k);
        v2f b = *(const v2f*)(wr + k);
        acc = __builtin_amdgcn_wmma_f32_16x16x4_f32(
            /*neg_a=*/false, a, /*neg_b=*/false, b,
            /*c_mod=*/(short)0, acc, /*reuse_a=*/false, /*reuse_b=*/false);
    }
    // neighbor term: (agg * inv) @ Wl^T  (mean fused into A-load)
#pragma unroll 8
    for (int k = 0; k < K; k += 4) {
        v2f a = *(const v2f*)(arow + k);
        a.x *= s;
        a.y *= s;
        v2f b = *(const v2f*)(wl + k);
        acc = __builtin_amdgcn_wmma_f32_16x16x4_f32(
            false, a, false, b, (short)0, acc, false, false);
    }

    const float bv = bias[col];
    const int mBase = m0 + ((lane >> 4) << 3);
#pragma unroll
    for (int r = 0; r < 8; ++r) {
        float v = acc[r] + bv;
        if (RELU) v = fmaxf(v, 0.0f);
        out[(size_t)(mBase + r) * Cout + col] = v;
    }
}

// ---------------- host-side launch ----------------
extern "C" void kernel_launch(void* const* d_in, const int* in_sizes, int n_in,
                              void* d_out, int out_size, void* d_ws, size_t ws_size,
                              hipStream_t stream) {
    const float* x   = (const float*)d_in[0];
    const int*   ei  = (const int*)d_in[1];   // [2, E]
    const float* W1l = (const float*)d_in[2];
    const float* W1r = (const float*)d_in[3];
    const float* b1  = (const float*)d_in[4];
    const float* W2l = (const float*)d_in[5];
    const float* W2r = (const float*)d_in[6];
    const float* b2  = (const float*)d_in[7];
    const float* W3l = (const float*)d_in[8];
    const float* W3r = (const float*)d_in[9];
    const float* b3  = (const float*)d_in[10];

    const int IN_CH = 128, HID_CH = 128, OUT_CH = 64;
    const int N = in_sizes[0] / IN_CH;   // 100000 (divisible by 16)
    const int E = in_sizes[1] / 2;       // 1600000

    const int* src = ei;
    const int* dst = ei + E;

    // workspace layout (~155 MB)
    float* h1  = (float*)d_ws;              // N*128
    float* h2  = h1 + (size_t)N * HID_CH;   // N*128
    float* agg = h2 + (size_t)N * HID_CH;   // N*128
    float* inv = agg + (size_t)N * HID_CH;  // N
    int*   cnt = (int*)(inv + N);           // N

    const long nAgg = (long)N * HID_CH;
    const int zBlocks = (int)((nAgg + 255) / 256);
    const int eBlocks = (E + 255) / 256;
    const int nBlocks = (N + 255) / 256;
    const int sBlocks = (int)(((long)E * 32 + 255) / 256);

    // degrees -> inverse degree
    fill_zero_i32<<<nBlocks, 256, 0, stream>>>(cnt, N);
    degree_kernel<<<eBlocks, 256, 0, stream>>>(dst, cnt, E);
    invdeg_kernel<<<nBlocks, 256, 0, stream>>>(cnt, inv, N);

    // ---- layer 1: x -> h1 (relu) ----
    fill_zero_f32<<<zBlocks, 256, 0, stream>>>(agg, nAgg);
    scatter_add128<<<sBlocks, 256, 0, stream>>>(x, src, dst, agg, E);
    sage_gemm_wmma<true><<<dim3(N / 16, HID_CH / 64), 128, 0, stream>>>(
        x, agg, inv, W1l, W1r, b1, h1, IN_CH, HID_CH);

    // ---- layer 2: h1 -> h2 (relu) ----
    fill_zero_f32<<<zBlocks, 256, 0, stream>>>(agg, nAgg);
    scatter_add128<<<sBlocks, 256, 0, stream>>>(h1, src, dst, agg, E);
    sage_gemm_wmma<true><<<dim3(N / 16, HID_CH / 64), 128, 0, stream>>>(
        h1, agg, inv, W2l, W2r, b2, h2, HID_CH, HID_CH);

    // ---- layer 3: h2 -> out (no relu, Cout = 64 -> grid.y = 1) ----
    fill_zero_f32<<<zBlocks, 256, 0, stream>>>(agg, nAgg);
    scatter_add128<<<sBlocks, 256, 0, stream>>>(h2, src, dst, agg, E);
    sage_gemm_wmma<false><<<dim3(N / 16, 1), 128, 0, stream>>>(
        h2, agg, inv, W3l, W3r, b3, (float*)d_out, HID_CH, OUT_CH);
}